// VQVAE_87479893885068
// MI455X (gfx1250) — compile-verified
//
#include <hip/hip_runtime.h>
#include <hip/hip_bf16.h>

// VQVAE quantize for MI455X (gfx1250, wave32).
// Distance GEMM on V_WMMA_F32_16X16X4_F32 (exact f32 math -> argmin matches ref).
// d_out layout: [quant 8*64*64*64][indices(as f32) 8*64*64][commit][codebook]

typedef __attribute__((ext_vector_type(2))) float v2f;
typedef __attribute__((ext_vector_type(8))) float v8f;

#define N_PIX   32768      // B*H*W = 8*4096
#define N_CODE  8192
#define N_CH    64
#define HWSZ    4096
#define ESTRIDE 68         // LDS row stride in floats: 16B aligned, bank-conflict-free

__global__ void vq_zero_losses(float* loss) {
    if (threadIdx.x < 2) loss[threadIdx.x] = 0.0f;
}

__global__ void vq_esq(const float* __restrict__ emb, float* __restrict__ esq) {
    int r = blockIdx.x * blockDim.x + threadIdx.x;
    if (r < N_CODE) {
        const float4* row = (const float4*)(emb + (size_t)r * N_CH);
        float s = 0.0f;
#pragma unroll
        for (int i = 0; i < N_CH / 4; ++i) {
            float4 v = row[i];
            s += v.x * v.x + v.y * v.y + v.z * v.z + v.w * v.w;
        }
        esq[r] = s;
    }
}

__global__ __launch_bounds__(256)
void vq_main(const float* __restrict__ x, const float* __restrict__ emb,
             const float* __restrict__ esq,
             float* __restrict__ out_q, float* __restrict__ out_idx,
             float* __restrict__ out_loss) {
    __shared__ float ebuf[2][16 * ESTRIDE];   // double-buffered 16-code tile
    __shared__ int   idx_s[256];              // per-pixel winning code

    const int tid  = threadIdx.x;
    const int lane = tid & 31;            // wave32
    const int w    = tid >> 5;            // wave id in block (0..7)
    const int ml   = lane & 15;           // row/col within 16
    const int half = lane >> 4;           // K-pair select per ISA A/B layout
    const int pbase = blockIdx.x * 256 + w * 32;   // 32 pixels per wave

    // ---- Load A fragments once: 2 M-tiles x 16 K-steps (K=4 each) ----
    // A 16x4 f32 layout: lanes 0-15 -> M rows; VGPR0={K0,K2}, VGPR1={K1,K3}.
    v2f a0[16], a1[16];
    {
        const int p0 = pbase + ml, p1 = pbase + 16 + ml;
        const int b0 = p0 >> 12, h0 = p0 & (HWSZ - 1);
        const int b1 = p1 >> 12, h1 = p1 & (HWSZ - 1);
#pragma unroll
        for (int kt = 0; kt < 16; ++kt) {
            const int c = kt * 4 + half * 2;
            a0[kt].x = x[((size_t)(b0 * N_CH + c    )) * HWSZ + h0];
            a0[kt].y = x[((size_t)(b0 * N_CH + c + 1)) * HWSZ + h0];
            a1[kt].x = x[((size_t)(b1 * N_CH + c    )) * HWSZ + h1];
            a1[kt].y = x[((size_t)(b1 * N_CH + c + 1)) * HWSZ + h1];
        }
    }

    float minv0[8], minv1[8];
    int   mini0[8], mini1[8];
#pragma unroll
    for (int v = 0; v < 8; ++v) {
        minv0[v] = 3.402823466e38f; minv1[v] = 3.402823466e38f;
        mini0[v] = 0; mini1[v] = 0;
    }

    // Cooperative tile loader: 256 threads x float4 = 1024 floats (16 codes x 64 ch)
    const int lr = tid >> 4;              // code row 0..15
    const int lc = (tid & 15) * 4;        // channel quad
    *(float4*)&ebuf[0][lr * ESTRIDE + lc] =
        *(const float4*)(emb + (size_t)lr * N_CH + lc);
    __syncthreads();

    for (int t = 0; t < N_CODE / 16; ++t) {
        const int cur = t & 1;
        if (t + 1 < N_CODE / 16) {
            *(float4*)&ebuf[cur ^ 1][lr * ESTRIDE + lc] =
                *(const float4*)(emb + ((size_t)(t + 1) * 16 + lr) * N_CH + lc);
            if (t + 2 < N_CODE / 16)
                __builtin_prefetch(emb + (size_t)(t + 2) * 16 * N_CH, 0, 0);
        }
        const float* eb = ebuf[cur];

        v8f acc0 = {}; v8f acc1 = {};
#pragma unroll
        for (int kt = 0; kt < 16; ++kt) {
            const int c = kt * 4 + half * 2;
            // B 4x16 layout: lanes 0-15 -> N cols; VGPR0={K0,K2}, VGPR1={K1,K3}
            v2f bf = *(const v2f*)&eb[ml * ESTRIDE + c];
            acc0 = __builtin_amdgcn_wmma_f32_16x16x4_f32(
                false, a0[kt], false, bf, (short)0, acc0, false, false);
            acc1 = __builtin_amdgcn_wmma_f32_16x16x4_f32(
                false, a1[kt], false, bf, (short)0, acc1, false, false);
        }

        const int code = t * 16 + ml;     // this lane's code column
        const float es = esq[code];
#pragma unroll
        for (int v = 0; v < 8; ++v) {
            const float s0 = fmaf(-2.0f, acc0[v], es);
            if (s0 < minv0[v]) { minv0[v] = s0; mini0[v] = code; }
            const float s1 = fmaf(-2.0f, acc1[v], es);
            if (s1 < minv1[v]) { minv1[v] = s1; mini1[v] = code; }
        }
        __syncthreads();   // buf cur free for reload; buf cur^1 load visible
    }

    // ---- Cross-lane argmin within each 16-lane half (ties -> lower index) ----
#pragma unroll
    for (int v = 0; v < 8; ++v) {
#pragma unroll
        for (int m = 8; m >= 1; m >>= 1) {
            float ov = __shfl_xor(minv0[v], m, 32);
            int   oi = __shfl_xor(mini0[v], m, 32);
            if (ov < minv0[v] || (ov == minv0[v] && oi < mini0[v])) {
                minv0[v] = ov; mini0[v] = oi;
            }
            ov = __shfl_xor(minv1[v], m, 32);
            oi = __shfl_xor(mini1[v], m, 32);
            if (ov < minv1[v] || (ov == minv1[v] && oi < mini1[v])) {
                minv1[v] = ov; mini1[v] = oi;
            }
        }
    }
    // C/D layout: lanes 0-15 -> M=v, lanes 16-31 -> M=v+8.
    if (ml == 0) {
#pragma unroll
        for (int v = 0; v < 8; ++v) {
            idx_s[w * 32 +      half * 8 + v] = mini0[v];
            idx_s[w * 32 + 16 + half * 8 + v] = mini1[v];
        }
    }
    __syncthreads();

    // ---- Outputs: one pixel per lane (coalesced quant stores per channel) ----
    const int p   = pbase + lane;
    const int idx = idx_s[w * 32 + lane];
    const int bi  = p >> 12;
    const int hw  = p & (HWSZ - 1);
    out_idx[p] = (float)idx;

    const float* erow = emb + (size_t)idx * N_CH;
    float sum = 0.0f;
#pragma unroll 8
    for (int c = 0; c < N_CH; ++c) {
        const float  q = erow[c];
        const size_t o = ((size_t)(bi * N_CH + c)) * HWSZ + hw;
        const float  d = x[o] - q;
        out_q[o] = q;                       // straight-through fwd value == quant
        sum = fmaf(d, d, sum);
    }
#pragma unroll
    for (int m = 16; m >= 1; m >>= 1) sum += __shfl_xor(sum, m, 32);
    if (lane == 0) {
        const float contrib = sum * (1.0f / ((float)N_PIX * (float)N_CH));
        atomicAdd(&out_loss[0], contrib);   // commitment_loss
        atomicAdd(&out_loss[1], contrib);   // codebook_loss (source bug: same value)
    }
}

extern "C" void kernel_launch(void* const* d_in, const int* in_sizes, int n_in,
                              void* d_out, int out_size, void* d_ws, size_t ws_size,
                              hipStream_t stream) {
    const float* x   = (const float*)d_in[0];
    const float* emb = (const float*)d_in[1];

    float* out      = (float*)d_out;
    float* out_q    = out;                       // 2,097,152 floats
    float* out_idx  = out + N_PIX * N_CH;        // 32,768 floats
    float* out_loss = out_idx + N_PIX;           // 2 floats
    float* esq      = (float*)d_ws;              // 8192 floats scratch

    vq_zero_losses<<<1, 32, 0, stream>>>(out_loss);
    vq_esq<<<N_CODE / 256, 256, 0, stream>>>(emb, esq);
    vq_main<<<N_PIX / 256, 256, 0, stream>>>(x, emb, esq, out_q, out_idx, out_loss);
}